// NarrativeIdentity_23270132810240
// MI455X (gfx1250) — compile-verified
//
#include <hip/hip_runtime.h>
#include <stdint.h>

typedef float v2f __attribute__((ext_vector_type(2)));
typedef float v4f __attribute__((ext_vector_type(4)));
typedef float v8f __attribute__((ext_vector_type(8)));

#define NBLK1            2048
#define TPB              256
#define WAVES_PER_BLOCK  (TPB / 32)
#define NSTREAM          6
#define TILE_FLOATS      128                 // 32 lanes * 4 floats per stream per tile
#define TILE_BYTES       (TILE_FLOATS * 4)   // 512 B

// ---- CDNA5 async global->LDS load (ASYNCcnt-tracked), GVS addressing ----
__device__ __forceinline__ void async_load_b128(uint32_t lds_off, uint32_t goff,
                                                const float* base) {
  asm volatile("global_load_async_to_lds_b128 %0, %1, %2"
               :: "v"(lds_off), "v"(goff), "s"(base)
               : "memory");
}

__global__ __launch_bounds__(TPB)
void ni_main_kernel(const float* __restrict__ css,   // 4*dim
                    const float* __restrict__ narr,  // dim
                    const float* __restrict__ tonic, // dim
                    float* __restrict__ out,         // 2*dim + 3
                    float* __restrict__ ws,          // 2 floats per block
                    int dim) {
  // Per-wave double-buffered LDS staging: 8 waves * 2 bufs * 6 streams * 512B = 48KB
  __shared__ __align__(16) float sm[WAVES_PER_BLOCK * 2 * NSTREAM * TILE_FLOATS];
  __shared__ float red[WAVES_PER_BLOCK * 2];

  const int w       = threadIdx.x >> 5;
  const int lane    = threadIdx.x & 31;
  const int ntiles  = dim / TILE_FLOATS;
  const int wglobal = blockIdx.x * WAVES_PER_BLOCK + w;
  const int wstride = gridDim.x * WAVES_PER_BLOCK;

  const float* base[NSTREAM] = { css,
                                 css + (size_t)dim,
                                 css + 2 * (size_t)dim,
                                 css + 3 * (size_t)dim,
                                 narr,
                                 tonic };

  const uint32_t lds0 = (uint32_t)(uintptr_t)sm;  // low 32 bits of generic ptr = LDS byte offset

  auto issue = [&](int t, int b) {
    const uint32_t goff = (uint32_t)t * (uint32_t)TILE_BYTES + (uint32_t)lane * 16u;
#pragma unroll
    for (int s = 0; s < NSTREAM; ++s) {
      const uint32_t loff =
          lds0 + (uint32_t)(((((w * 2 + b) * NSTREAM) + s) * TILE_FLOATS + lane * 4) * 4);
      async_load_b128(loff, goff, base[s]);
    }
  };

  float accd = 0.0f;  // sum of deviation^2
  float accc = 0.0f;  // sum of css^2

  int buf = 0;
  int t = wglobal;
  if (t < ntiles) issue(t, 0);  // prologue: prefetch first tile

  for (; t < ntiles; t += wstride) {
    const int  tn = t + wstride;
    const bool hn = (tn < ntiles);
    if (hn) {
      issue(tn, buf ^ 1);                                   // prefetch next tile
      asm volatile("s_wait_asynccnt 0x6" ::: "memory");     // oldest 6 (this tile) done
    } else {
      asm volatile("s_wait_asynccnt 0x0" ::: "memory");
    }

    const float* p = &sm[((w * 2 + buf) * NSTREAM) * TILE_FLOATS + lane * 4];
    const v4f a0 = *(const v4f*)(p + 0 * TILE_FLOATS);
    const v4f a1 = *(const v4f*)(p + 1 * TILE_FLOATS);
    const v4f a2 = *(const v4f*)(p + 2 * TILE_FLOATS);
    const v4f a3 = *(const v4f*)(p + 3 * TILE_FLOATS);
    const v4f nv = *(const v4f*)(p + 4 * TILE_FLOATS);
    const v4f tv = *(const v4f*)(p + 5 * TILE_FLOATS);

    const v4f cs = 0.25f * (a0 + a1 + a2 + a3);             // batch mean
    const v4f dv = cs - nv;                                  // deviation
    accd += dv.x * dv.x + dv.y * dv.y + dv.z * dv.z + dv.w * dv.w;
    accc += cs.x * cs.x + cs.y * cs.y + cs.z * cs.z + cs.w * cs.w;

    const v4f nn = nv * 0.99f + cs * 0.01f + tv;             // EMA + tonic drift
    const v4f ve = nn - nv;                                  // velocity

    const size_t e0 = (size_t)t * TILE_FLOATS + (size_t)lane * 4;
    __builtin_nontemporal_store(nn, (v4f*)(out + e0));
    __builtin_nontemporal_store(ve, (v4f*)(out + (size_t)dim + e0));

    buf ^= 1;
  }

  // deterministic wave reduction (fixed xor tree)
#pragma unroll
  for (int off = 16; off > 0; off >>= 1) {
    accd += __shfl_xor(accd, off, 32);
    accc += __shfl_xor(accc, off, 32);
  }
  if (lane == 0) { red[w * 2] = accd; red[w * 2 + 1] = accc; }
  __syncthreads();
  if (threadIdx.x == 0) {
    float sd = 0.0f, sc = 0.0f;
#pragma unroll
    for (int i = 0; i < WAVES_PER_BLOCK; ++i) { sd += red[2 * i]; sc += red[2 * i + 1]; }
    ws[(size_t)blockIdx.x * 2]     = sd;
    ws[(size_t)blockIdx.x * 2 + 1] = sc;
  }
}

// Deterministic 32-lane sum using V_WMMA_F32_16X16X4_F32 with a ones B-matrix:
// A VGPR0 carries lane partials (K=0 for lanes 0-15, K=2 for lanes 16-31), VGPR1=0.
// D[m][n] = p_m + p_{m+16}; summing D's 8 regs per lane and combining lanes 0 & 16
// yields the full 32-lane sum in a fixed order.
__device__ __forceinline__ float wave32_sum(float x) {
#if __has_builtin(__builtin_amdgcn_wmma_f32_16x16x4_f32)
  v2f A; A.x = x;    A.y = 0.0f;
  v2f B; B.x = 1.0f; B.y = 1.0f;
  v8f C = {};
  v8f D = __builtin_amdgcn_wmma_f32_16x16x4_f32(false, A, false, B, (short)0, C,
                                                false, false);
  float t = D[0] + D[1] + D[2] + D[3] + D[4] + D[5] + D[6] + D[7];
  return __shfl(t, 0, 32) + __shfl(t, 16, 32);
#else
#pragma unroll
  for (int off = 16; off > 0; off >>= 1) x += __shfl_xor(x, off, 32);
  return x;
#endif
}

__global__ __launch_bounds__(32)
void ni_finalize_kernel(const float* __restrict__ ws,
                        const float* __restrict__ idfp,
                        const float* __restrict__ idsp,
                        float* __restrict__ out,
                        int nblk, int dim) {
  const int lane = threadIdx.x;
  float sd = 0.0f, sc = 0.0f;
  for (int i = lane; i < nblk; i += 32) { sd += ws[2 * i]; sc += ws[2 * i + 1]; }

  const float sum_dev2 = wave32_sum(sd);
  const float sum_css2 = wave32_sum(sc);

  if (lane == 0) {
    const float dev_norm = sqrtf(sum_dev2);
    const float css_norm = sqrtf(sum_css2);
    const float consistency_loss = 0.1f * dev_norm;

    const float gate = fminf(1.0f, css_norm);
    const float idf  = idfp[0];
    const float ids  = idsp[0];

    const float delta_fast = 0.01f / (1.0f + idf * 5.0f) * gate;
    const float fast_new   = fminf(idf * 0.9995f + delta_fast, 0.7f);

    const float fit        = fminf(fmaxf(1.0f - dev_norm, 0.0f), 1.0f);
    const float mult       = 1.0f + 15.0f * fit * fit;
    const float delta_slow = 0.001f / (1.0f + ids * 5.0f) * mult;
    const float slow_new   = fminf(ids + delta_slow, 1.0f);

    const float strength = fminf(fast_new + slow_new, 1.0f);

    out[(size_t)2 * dim + 0] = consistency_loss;
    out[(size_t)2 * dim + 1] = strength;
    out[(size_t)2 * dim + 2] = dev_norm;
  }
}

extern "C" void kernel_launch(void* const* d_in, const int* in_sizes, int n_in,
                              void* d_out, int out_size, void* d_ws, size_t ws_size,
                              hipStream_t stream) {
  const float* css   = (const float*)d_in[0];  // (4, DIM)
  const float* narr  = (const float*)d_in[1];  // (DIM,)
  const float* tonic = (const float*)d_in[2];  // (DIM,)
  const float* idf   = (const float*)d_in[3];  // scalar
  const float* ids   = (const float*)d_in[4];  // scalar
  float* out = (float*)d_out;
  float* ws  = (float*)d_ws;
  const int dim = in_sizes[1];

  int nblk = NBLK1;
  if (ws_size < (size_t)nblk * 2 * sizeof(float)) {
    nblk = (int)(ws_size / (2 * sizeof(float)));
    if (nblk < 1) nblk = 1;
  }

  ni_main_kernel<<<nblk, TPB, 0, stream>>>(css, narr, tonic, out, ws, dim);
  ni_finalize_kernel<<<1, 32, 0, stream>>>(ws, idf, ids, out, nblk, dim);
}